// HybridContrastiveLoss_25348896981165
// MI455X (gfx1250) — compile-verified
//
#include <hip/hip_runtime.h>

// HybridContrastiveLoss for MI455X (gfx1250, wave32).
// ws layout: [0,16MB)   fnT   = l2-normalized features, channels-last [n][i][j][c]
//            [16MB, +4KB)  pDir  = 1024 directional block partials
//            [+4KB, +64KB) pLoc  = 16384 local wave partials

typedef float v2f __attribute__((ext_vector_type(2)));
typedef float v8f __attribute__((ext_vector_type(8)));

#define N_ 4
#define C_ 16
#define H_ 256
#define W_ 256
#define HW_ (H_ * W_)
#define NS_ 5
#define NOFF 121          // (2*NS+1)^2
#define INV_TEMP 10.0f
#define EPS_ 1e-6f

// ---------------------------------------------------------------- normalize
// One thread per pixel: read 16 channel-strided floats, l2-normalize,
// write 16 contiguous floats (channels-last) -> enables float2 WMMA fragments.
__global__ void hcl_normalize(const float* __restrict__ feat,
                              float* __restrict__ fnT) {
  int t = blockIdx.x * 256 + threadIdx.x;      // pixel id, exact grid
  int n = t >> 16;
  int hw = t & 0xFFFF;
  const float* src = feat + (size_t)n * C_ * HW_ + hw;
  float v[C_];
  float ss = 0.f;
#pragma unroll
  for (int c = 0; c < C_; ++c) {
    v[c] = src[(size_t)c * HW_];
    ss += v[c] * v[c];
  }
  float inv = 1.0f / fmaxf(sqrtf(ss), 1e-12f);
  float* dst = fnT + (size_t)t * C_;
#pragma unroll
  for (int c = 0; c < C_; ++c) dst[c] = v[c] * inv;
}

// ---------------------------------------------------------------- directional
// One thread per (a,i,j): 4 gathered 16-dim dots, masked softmax term.
// Block-tree reduction (deterministic) -> pDir[block].
__global__ void hcl_directional(const float* __restrict__ fnT,
                                const int* __restrict__ labels,
                                const int* __restrict__ dirs,
                                float* __restrict__ pDir) {
  __shared__ float red[256];
  int t = blockIdx.x * 256 + threadIdx.x;      // exact: 1024*256 = N*H*W
  int a = t >> 16;
  int hw = t & 0xFFFF;
  int i = hw >> 8;
  int j = hw & 255;
  const float* fc = fnT + (size_t)t * C_;
  int labc[4];
  float e[4];
  float S = 0.f;
#pragma unroll
  for (int k = 0; k < 4; ++k) {
    int di = dirs[((size_t)(k * 2 + 0) * H_ + i) * W_ + j];
    int dj = dirs[((size_t)(k * 2 + 1) * H_ + i) * W_ + j];
    int ni = i + di;                           // pre-clipped in setup
    int nj = j + dj;
    const float* fb = fnT + ((size_t)(a * H_ + ni) * W_ + nj) * C_;
    float d = 0.f;
#pragma unroll
    for (int c = 0; c < C_; ++c) d += fc[c] * fb[c];
    float l = d * INV_TEMP;
    labc[k] = (labels[k * HW_ + hw] == labels[a * HW_ + ni * W_ + nj]);
    e[k] = labc[k] ? expf(l) : 0.0f;
    S += e[k];
  }
  float ls = logf(S + EPS_);
  float tsum = 0.f;
#pragma unroll
  for (int k = 0; k < 4; ++k) tsum += ls - logf(e[k]);

  red[threadIdx.x] = tsum;
  __syncthreads();
  for (int s = 128; s > 0; s >>= 1) {
    if (threadIdx.x < s) red[threadIdx.x] += red[threadIdx.x + s];
    __syncthreads();
  }
  if (threadIdx.x == 0) pDir[blockIdx.x] = red[0];
}

// ---------------------------------------------------------------- local (WMMA)
// One wave per (n, i, jtile of 16 center pixels).
// Phase 1: for each valid row offset p, compute the 3 overlapping 16x16 tiles
//   of the Gram matrix rowI x rowI2 with V_WMMA_F32_16X16X4_F32 (K=16 via 4
//   chained WMMAs, f32 precision), scatter the |q|<=5 band into LDS.
// Phase 2: lanes 0..15 each do the masked log-sum-exp over 121 offsets for
//   their pixel; wave shuffle-reduce -> pLoc[wave].
#define WPB 4  // waves per block
__global__ void hcl_local(const float* __restrict__ fnT,
                          const int* __restrict__ labels,
                          float* __restrict__ pLoc) {
  __shared__ float smem[WPB][16][128];         // [wave][center px][offset] 32KB
  int wid = threadIdx.x >> 5;
  int lane = threadIdx.x & 31;
  int gw = blockIdx.x * WPB + wid;             // 0..16383
  int jt = gw & 15;
  int i = (gw >> 4) & 255;
  int n = gw >> 12;
  int jA = jt * 16;

  const float* rowC = fnT + ((size_t)(n * H_ + i) * W_) * C_;
  int pixM = lane & 15;                        // A: M index / B: N index
  int cbase = (lane >> 4) * 2;                 // K split across lane halves

  // A fragments (center row), reused across all 11 row offsets.
  v2f afrag[4];
#pragma unroll
  for (int kk = 0; kk < 4; ++kk)
    afrag[kk] = *(const v2f*)(rowC + (jA + pixM) * C_ + cbase + 4 * kk);

  for (int p = -NS_; p <= NS_; ++p) {          // wave-uniform branch
    int i2 = i + p;
    if (i2 < 0 || i2 >= H_) continue;
    const float* rowN = fnT + ((size_t)(n * H_ + i2) * W_) * C_;
#pragma unroll
    for (int tb = 0; tb < 3; ++tb) {
      int jB = jA + (tb - 1) * 16;
      int pixB = jB + pixM;
      int pixBc = pixB < 0 ? 0 : (pixB > W_ - 1 ? W_ - 1 : pixB);
      v8f acc = {0.f, 0.f, 0.f, 0.f, 0.f, 0.f, 0.f, 0.f};
#pragma unroll
      for (int kk = 0; kk < 4; ++kk) {
        v2f b = *(const v2f*)(rowN + pixBc * C_ + cbase + 4 * kk);
        acc = __builtin_amdgcn_wmma_f32_16x16x4_f32(
            false, afrag[kk], false, b, (short)0, acc, false, false);
      }
      // D layout: lane holds column N = lane&15, rows M = v + 8*(lane>>4)
#pragma unroll
      for (int v = 0; v < 8; ++v) {
        int M = v + ((lane >> 4) << 3);
        int Nn = lane & 15;
        int q = (jB + Nn) - (jA + M);
        if (q >= -NS_ && q <= NS_)
          smem[wid][M][(p + NS_) * 11 + (q + NS_)] = acc[v] * INV_TEMP;
      }
    }
  }

  // Phase 2: per-pixel masked log-sum-exp over 121 offsets.
  float contrib = 0.f;
  if (lane < 16) {
    int j = jA + lane;
    int labc = labels[(n * H_ + i) * W_ + j];
    float S = 0.f, A1 = 0.f;
    int cntM = 0, cntV = 0;
    for (int o = 0; o < NOFF; ++o) {
      int p = o / 11 - NS_;
      int q = o % 11 - NS_;
      int i2 = i + p, j2 = j + q;
      if (i2 < 0 || i2 >= H_ || j2 < 0 || j2 >= W_) continue;
      ++cntV;
      if (labels[(n * H_ + i2) * W_ + j2] == labc) {
        float l = smem[wid][lane][o];
        S += expf(l);
        A1 += l;
        ++cntM;
      }
    }
    // sum_o term*m*v = cntM*log(S+eps) - A1 ; final scale /(N*count*H*W)
    contrib = ((float)cntM * logf(S + EPS_) - A1) /
              (4.0f * (float)cntV * (float)HW_);
  }
#pragma unroll
  for (int off = 16; off > 0; off >>= 1)
    contrib += __shfl_down(contrib, off, 32);
  if (lane == 0) pLoc[gw] = contrib;
}

// ---------------------------------------------------------------- final reduce
// Single block, fixed strided + tree order -> deterministic scalar output.
__global__ void hcl_finalize(const float* __restrict__ pDir,
                             const float* __restrict__ pLoc,
                             float* __restrict__ out) {
  __shared__ float red[256];
  float sd = 0.f;
  for (int idx = threadIdx.x; idx < 1024; idx += 256) sd += pDir[idx];
  float sl = 0.f;
  for (int idx = threadIdx.x; idx < 16384; idx += 256) sl += pLoc[idx];
  // directional mean over N*4*H*W = 1048576 elements
  red[threadIdx.x] = sd * (1.0f / 1048576.0f) + sl;
  __syncthreads();
  for (int s = 128; s > 0; s >>= 1) {
    if (threadIdx.x < s) red[threadIdx.x] += red[threadIdx.x + s];
    __syncthreads();
  }
  if (threadIdx.x == 0) out[0] = red[0];
}

// ---------------------------------------------------------------- launcher
extern "C" void kernel_launch(void* const* d_in, const int* in_sizes, int n_in,
                              void* d_out, int out_size, void* d_ws,
                              size_t ws_size, hipStream_t stream) {
  const float* features = (const float*)d_in[0];  // [4,16,256,256] f32
  const int* labels = (const int*)d_in[1];        // [4,256,256] i32
  const int* dirs = (const int*)d_in[2];          // [4,2,256,256] i32

  float* fnT = (float*)d_ws;                                   // 16 MB
  float* pDir = (float*)((char*)d_ws + (size_t)16777216);      // 4 KB
  float* pLoc = pDir + 1024;                                   // 64 KB

  hcl_normalize<<<1024, 256, 0, stream>>>(features, fnT);
  hcl_directional<<<1024, 256, 0, stream>>>(fnT, labels, dirs, pDir);
  hcl_local<<<4096, 32 * WPB, 0, stream>>>(fnT, labels, pLoc);
  hcl_finalize<<<1, 256, 0, stream>>>(pDir, pLoc, (float*)d_out);
}